// GCNII_model_17952963298032
// MI455X (gfx1250) — compile-verified
//
#include <hip/hip_runtime.h>
#include <math.h>

#define NN 100000
#define NE 1600000
#define NF 64
#define NLAYER 4
#define ALPHA_C 0.1f

typedef float v2f __attribute__((ext_vector_type(2)));
typedef float v8f __attribute__((ext_vector_type(8)));

// ---------------- small utility kernels ----------------

__global__ __launch_bounds__(256) void fill_zero_k(float* __restrict__ p, int n) {
    int i = blockIdx.x * 256 + threadIdx.x;
    if (i < n) p[i] = 0.0f;
}

__global__ __launch_bounds__(256) void degree_k(const int* __restrict__ col,
                                                float* __restrict__ deg, int ne) {
    int e = blockIdx.x * 256 + threadIdx.x;
    if (e < ne) unsafeAtomicAdd(&deg[col[e]], 1.0f);
}

// dis[i] = rsqrt(deg[i] + 1)   (the +1 is the self-loop)
__global__ __launch_bounds__(256) void dis_k(float* __restrict__ deg, int n) {
    int i = blockIdx.x * 256 + threadIdx.x;
    if (i < n) deg[i] = rsqrtf(deg[i] + 1.0f);
}

// ---------------- input projection: h = relu(x @ W_in + b_in), K = 17 ----------------
// 4 nodes x 64 cols per 256-thread block; W_in and the 4 x-rows staged in LDS.
__global__ __launch_bounds__(256) void inproj_k(const float* __restrict__ x,
                                                const float* __restrict__ Win,
                                                const float* __restrict__ bin,
                                                float* __restrict__ h,
                                                float* __restrict__ h0) {
    __shared__ float sW[17 * 64];
    __shared__ float sx[4 * 17];
    for (int i = threadIdx.x; i < 17 * 64; i += 256) sW[i] = Win[i];
    const int nodeBase = blockIdx.x * 4;
    for (int i = threadIdx.x; i < 4 * 17; i += 256) sx[i] = x[nodeBase * 17 + i];
    __syncthreads();

    const int local = threadIdx.x >> 6;       // 0..3
    const int col   = threadIdx.x & 63;
    const int node  = nodeBase + local;
    float acc = bin[col];
#pragma unroll
    for (int j = 0; j < 17; ++j)
        acc = fmaf(sx[local * 17 + j], sW[j * 64 + col], acc);
    float v = fmaxf(acc, 0.0f);
    h [node * NF + col] = v;
    h0[node * NF + col] = v;
}

// ---------------- dual fp32 WMMA GEMM ----------------
// Block = 8 waves, one 16-node row block.
//   waves 0..3 : support = s_h  * h  + beta * (h  @ W1)   (col tiles 0..3)
//   waves 4..7 : nxt     = s_h0 * h0 + beta * (h0 @ W2)   (col tiles 0..3)
// Each wave chains 16 x V_WMMA_F32_16X16X4_F32 over K=64.
__global__ __launch_bounds__(256) void dual_gemm_k(const float* __restrict__ h,
                                                   const float* __restrict__ h0,
                                                   const float* __restrict__ W1,
                                                   const float* __restrict__ W2,
                                                   float* __restrict__ support,
                                                   float* __restrict__ nxt,
                                                   float beta, float s_h, float s_h0) {
    const int wave = threadIdx.x >> 5;   // 0..7 (wave32)
    const int lane = threadIdx.x & 31;
    const int half = lane >> 4;          // 0 or 1
    const int l16  = lane & 15;
    const int R = blockIdx.x * 16;       // node row base (exact: 100000 = 6250*16)
    const int C = (wave & 3) * 16;       // output column base

    const bool doInit = (wave >= 4);     // uniform per wave -> EXEC stays all-1s
    const float* __restrict__ X = doInit ? h0 : h;
    const float* __restrict__ W = doInit ? W2 : W1;

    v8f acc = {};
#pragma unroll
    for (int kk = 0; kk < 16; ++kk) {
        const int kb = kk * 4 + half * 2;
        v2f a, b;
        // A fragment: 16x4 tile of X, row = R+l16, K pair selected by half
        a.x = X[(R + l16) * NF + kb + 0];
        a.y = X[(R + l16) * NF + kb + 1];
        // B fragment: 4x16 tile of W (row-major K x N), N = C + l16
        b.x = W[(kb + 0) * NF + C + l16];
        b.y = W[(kb + 1) * NF + C + l16];
        acc = __builtin_amdgcn_wmma_f32_16x16x4_f32(false, a, false, b,
                                                    (short)0, acc, false, false);
    }

    // C/D layout: VGPR r -> M = r + 8*half (lanes 0-15 / 16-31), N = C + l16
#pragma unroll
    for (int r = 0; r < 8; ++r) {
        const int node = R + half * 8 + r;
        const int idx  = node * NF + C + l16;
        const float d  = beta * acc[r];
        if (doInit) nxt[idx]     = fmaf(s_h0, h0[idx], d);
        else        support[idx] = fmaf(s_h,  h [idx], d);
    }
}

// ---------------- self-loop: nxt += dis[i]^2 * support[i] (no atomics needed) ----------------
__global__ __launch_bounds__(256) void selfloop_k(const float* __restrict__ support,
                                                  const float* __restrict__ dis,
                                                  float* __restrict__ nxt, int n) {
    int t = blockIdx.x * 256 + threadIdx.x;
    if (t < n) {
        int node = t >> 6;
        float w = dis[node] * dis[node];
        nxt[t] = fmaf(w, support[t], nxt[t]);
    }
}

// ---------------- edge scatter (SpMM): 16 lanes per edge, float4 gather + HW f32 atomics ----------------
__global__ __launch_bounds__(256) void spmm_k(const int* __restrict__ rows,
                                              const int* __restrict__ cols,
                                              const float* __restrict__ dis,
                                              const float* __restrict__ support,
                                              float* __restrict__ nxt, int ne) {
    long long t = (long long)blockIdx.x * 256 + threadIdx.x;
    int e = (int)(t >> 4);
    if (e >= ne) return;
    int fi = ((int)t & 15) * 4;
    int r = rows[e], c = cols[e];
    float w = dis[r] * dis[c];
    const float4 s = *(const float4*)(support + r * NF + fi);
    float* o = nxt + c * NF + fi;
    unsafeAtomicAdd(o + 0, w * s.x);
    unsafeAtomicAdd(o + 1, w * s.y);
    unsafeAtomicAdd(o + 2, w * s.z);
    unsafeAtomicAdd(o + 3, w * s.w);
}

// ---------------- relu: h = max(nxt, 0) ----------------
__global__ __launch_bounds__(256) void relu_k(const float* __restrict__ nxt,
                                              float* __restrict__ h, int n) {
    int t = blockIdx.x * 256 + threadIdx.x;
    if (t < n) h[t] = fmaxf(nxt[t], 0.0f);
}

// ---------------- output head: logits = h @ W_out + b_out, log_softmax over 2 classes ----------------
__global__ __launch_bounds__(256) void out_k(const float* __restrict__ h,
                                             const float* __restrict__ Wout,
                                             const float* __restrict__ bout,
                                             float* __restrict__ out, int n) {
    int node = blockIdx.x * 256 + threadIdx.x;
    if (node >= n) return;
    float l0 = bout[0], l1 = bout[1];
#pragma unroll
    for (int j = 0; j < NF; ++j) {
        float v = h[node * NF + j];
        l0 = fmaf(v, Wout[2 * j + 0], l0);
        l1 = fmaf(v, Wout[2 * j + 1], l1);
    }
    float m   = fmaxf(l0, l1);
    float lse = m + logf(expf(l0 - m) + expf(l1 - m));
    out[2 * node + 0] = l0 - lse;
    out[2 * node + 1] = l1 - lse;
}

// ---------------- launcher ----------------
extern "C" void kernel_launch(void* const* d_in, const int* in_sizes, int n_in,
                              void* d_out, int out_size, void* d_ws, size_t ws_size,
                              hipStream_t stream) {
    const float* x    = (const float*)d_in[0];
    const int*   ei   = (const int*)  d_in[1];   // [2, NE] flat: rows then cols
    const float* Win  = (const float*)d_in[2];
    const float* bin  = (const float*)d_in[3];
    const float* W1   = (const float*)d_in[4];   // [4,64,64]
    const float* W2   = (const float*)d_in[5];   // [4,64,64]
    const float* Wout = (const float*)d_in[6];
    const float* bout = (const float*)d_in[7];
    float* out = (float*)d_out;

    const int* rows = ei;
    const int* cols = ei + NE;

    float* ws      = (float*)d_ws;
    float* dis     = ws;                    // NN floats (deg, then dis in place)
    float* h0      = ws + (1 << 17);        // NN*NF
    float* h       = h0 + (size_t)NN * NF;  // NN*NF
    float* support = h  + (size_t)NN * NF;  // NN*NF
    float* nxt     = support + (size_t)NN * NF;

    // gcn_norm
    fill_zero_k<<<(NN + 255) / 256, 256, 0, stream>>>(dis, NN);
    degree_k  <<<(NE + 255) / 256, 256, 0, stream>>>(cols, dis, NE);
    dis_k     <<<(NN + 255) / 256, 256, 0, stream>>>(dis, NN);

    // input projection
    inproj_k<<<NN / 4, 256, 0, stream>>>(x, Win, bin, h, h0);

    for (int i = 0; i < NLAYER; ++i) {
        float beta = logf(0.5f / (float)(i + 1) + 1.0f);
        float s_h  = (1.0f - beta) * (1.0f - ALPHA_C);
        float s_h0 = (1.0f - beta) * ALPHA_C;
        dual_gemm_k<<<NN / 16, 256, 0, stream>>>(h, h0, W1 + (size_t)i * NF * NF,
                                                 W2 + (size_t)i * NF * NF,
                                                 support, nxt, beta, s_h, s_h0);
        selfloop_k<<<NN * NF / 256, 256, 0, stream>>>(support, dis, nxt, NN * NF);
        spmm_k    <<<NE * 16 / 256, 256, 0, stream>>>(rows, cols, dis, support, nxt, NE);
        relu_k    <<<NN * NF / 256, 256, 0, stream>>>(nxt, h, NN * NF);
    }

    out_k<<<(NN + 255) / 256, 256, 0, stream>>>(h, Wout, bout, out, NN);
}